// yolov1_loss_13494787244801
// MI455X (gfx1250) — compile-verified
//
#include <hip/hip_runtime.h>

// YOLOv1 loss on MI455X (gfx1250, wave32).
// Bandwidth-bound streaming reduction: ~385 MB read once -> scalar.
// NT b64 loads, per-thread cell loss, wave reduce via V_WMMA_F32_16X16X4_F32
// (ones-matrix trick), LDS block reduce, one atomicAdd per block.

#define S_GRID   14
#define BATCH_N  8192
#define NCELL    (BATCH_N * S_GRID * S_GRID)   // 1,605,632
#define CCH      30                             // 5*B + CLS
#define BLOCK    256

typedef float v2f __attribute__((ext_vector_type(2)));
typedef float v8f __attribute__((ext_vector_type(8)));

__device__ __forceinline__ float sq(float x) { return x * x; }

__device__ __forceinline__ float cell_loss(const float* __restrict__ pv,
                                           const float* __restrict__ tv) {
    // Decode both boxes of pred & target to corners, paired IoU per box slot.
    float iou[2];
#pragma unroll
    for (int b = 0; b < 2; ++b) {
        const float* pb = pv + 5 * b;
        const float* tb = tv + 5 * b;
        float phw = 0.5f * pb[2] * pb[2], phh = 0.5f * pb[3] * pb[3];
        float thw = 0.5f * tb[2] * tb[2], thh = 0.5f * tb[3] * tb[3];
        float px0 = pb[0] - phw, py0 = pb[1] - phh;
        float px1 = pb[0] + phw, py1 = pb[1] + phh;
        float tx0 = tb[0] - thw, ty0 = tb[1] - thh;
        float tx1 = tb[0] + thw, ty1 = tb[1] + thh;
        float lt0 = fmaxf(tx0, px0), lt1 = fmaxf(ty0, py0);
        float rb0 = fminf(tx1, px1), rb1 = fminf(ty1, py1);
        float w = fmaxf(rb0 - lt0, 0.0f), h = fmaxf(rb1 - lt1, 0.0f);
        float inter = w * h;
        float a1 = (tx1 - tx0) * (ty1 - ty0);
        float a2 = (px1 - px0) * (py1 - py0);
        float den = a1 + a2 - inter;
        iou[b] = (den > 0.0f) ? (inter / den) : 0.0f;
    }
    // argmax with first-wins tie break (matches jnp.argmax)
    int idx   = (iou[1] > iou[0]) ? 1 : 0;
    float miou = fmaxf(iou[0], iou[1]);

    const float* ps = pv + 5 * idx;
    const float* ts = tv + 5 * idx;

    // localization (xy + sqrt-wh), responsible box
    float local = sq(ps[0] - ts[0]) + sq(ps[1] - ts[1]) +
                  sq(ps[2] - ts[2]) + sq(ps[3] - ts[3]);
    // classification MSE over 20 classes
    float cls = 0.0f;
#pragma unroll
    for (int k = 10; k < 30; ++k) cls += sq(pv[k] - tv[k]);
    // containment: responsible pred conf vs detached max IoU
    float contain = sq(ps[4] - miou);
    // no-object: both box confidences in empty cells
    float noobj = sq(pv[4] - tv[4]) + sq(pv[9] - tv[9]);
    // iou loss: detached max IoU vs responsible target conf
    float iouL = sq(miou - ts[4]);

    float obj = (tv[4] > 0.0f) ? 1.0f : 0.0f;
    return obj * (5.0f * local + cls + contain + iouL) +
           (1.0f - obj) * (0.5f * noobj);
}

__global__ void __launch_bounds__(BLOCK)
yolov1_loss_kernel(const float* __restrict__ pred,
                   const float* __restrict__ target,
                   float* __restrict__ out) {
    __shared__ float smem[BLOCK / 32];

    float acc = 0.0f;
    const long long stride = (long long)gridDim.x * BLOCK;
    for (long long i = (long long)blockIdx.x * BLOCK + threadIdx.x;
         i < (long long)NCELL; i += stride) {
        // 30 contiguous floats per cell; cell base is 8B-aligned (120B stride)
        const v2f* p2 = (const v2f*)(pred   + (size_t)i * CCH);
        const v2f* t2 = (const v2f*)(target + (size_t)i * CCH);
        float pv[CCH], tv[CCH];
#pragma unroll
        for (int k = 0; k < 15; ++k) {
            v2f a = __builtin_nontemporal_load(p2 + k);
            v2f b = __builtin_nontemporal_load(t2 + k);
            pv[2 * k] = a.x; pv[2 * k + 1] = a.y;
            tv[2 * k] = b.x; tv[2 * k + 1] = b.y;
        }
        acc += cell_loss(pv, tv);
    }

    // ---- wave32 reduction via the matrix pipe ----
    // A = 16x4 ones; B carries one partial per lane (second B VGPR = 0).
    // D[m][n] = colsum(n) for every m, so summing D's VGPR0 over all 32 lanes
    // (rows M=0 and M=8) gives exactly 2x the wave total, for ANY B layout.
    v2f A;  A.x = 1.0f; A.y = 1.0f;
    v2f Bm; Bm.x = acc; Bm.y = 0.0f;
    v8f Cz = {};
    v8f D = __builtin_amdgcn_wmma_f32_16x16x4_f32(
        /*neg_a=*/false, A, /*neg_b=*/false, Bm,
        /*c_mod=*/(short)0, Cz, /*reuse_a=*/false, /*reuse_b=*/false);
    float r = D[0];
#pragma unroll
    for (int off = 16; off > 0; off >>= 1) r += __shfl_xor(r, off, 32);
    r *= 0.5f;  // rows M=0 and M=8 both summed

    const int lane = threadIdx.x & 31;
    const int wid  = threadIdx.x >> 5;
    if (lane == 0) smem[wid] = r;
    __syncthreads();

    if (wid == 0) {
        float v = (lane < (BLOCK / 32)) ? smem[lane] : 0.0f;
#pragma unroll
        for (int off = 4; off > 0; off >>= 1) v += __shfl_xor(v, off, 32);
        if (lane == 0)
            atomicAdd(out, v * (1.0f / (float)BATCH_N));
    }
}

__global__ void yolov1_zero_kernel(float* __restrict__ out) {
    if (threadIdx.x == 0 && blockIdx.x == 0) out[0] = 0.0f;
}

extern "C" void kernel_launch(void* const* d_in, const int* in_sizes, int n_in,
                              void* d_out, int out_size, void* d_ws, size_t ws_size,
                              hipStream_t stream) {
    (void)in_sizes; (void)n_in; (void)d_ws; (void)ws_size; (void)out_size;
    const float* pred   = (const float*)d_in[0];
    const float* target = (const float*)d_in[1];
    float* out = (float*)d_out;

    yolov1_zero_kernel<<<1, 32, 0, stream>>>(out);

    const int blocks = (NCELL + BLOCK - 1) / BLOCK;  // 6272, exact cover
    yolov1_loss_kernel<<<blocks, BLOCK, 0, stream>>>(pred, target, out);
}